// KnowledgeCircuit_3092376453536
// MI455X (gfx1250) — compile-verified
//
#include <hip/hip_runtime.h>
#include <stdint.h>

// KnowledgeCircuit on MI455X (gfx1250, wave32, WMMA + async-to-LDS).
//
// h   = einsum('bsd,ndr,bsn->bsr') : 32x GEMM (4096x1024)@(1024x128), row-scaled combine
// out = einsum('bsr,nrd,bsn->bsd') : 32x GEMM (4096x128)@(128x1024), row-scaled combine
//
// ~68.7 GFLOP vs ~70MB unique bytes (all L2-resident: 192MB) -> compute bound.
// fp32 matrix path = V_WMMA_F32_16X16X4_F32 (matches fp32 reference).
// Expert weights folded into the A fragment (per-row scale) so all 32 experts
// accumulate directly in the WMMA accumulators.
//
// Data movement: B chunks (and phase-1 A chunks) are streamed into LDS with
// GLOBAL_LOAD_ASYNC_TO_LDS_B128 (ASYNCcnt), double-buffered so the next
// expert's tile lands while WMMAs consume the current one. MT=64 + split-K
// (gridDim.z=2, partial h buffers summed deterministically in phase 2)
// halves L2 traffic while keeping >=128 blocks in flight per phase.

typedef __attribute__((ext_vector_type(2))) float v2f;
typedef __attribute__((ext_vector_type(8))) float v8f;

#define NE   32          // experts
#define MT   64          // token rows per block
#define KC   32          // K chunk staged in LDS
#define CT   128         // output columns per block
#define ASTR 36          // sA row stride (dwords): 16B-aligned, conflict-free b64 reads
#define WSTR 33          // sW row stride (dwords): conflict-free scalar reads
#define BSTR 136         // sB row stride (dwords): 2*136 % 64 == 16 -> conflict-free pairs
#define BBUF (KC * BSTR) // one B buffer: 4352 dwords (17408 B)

// Low 32 bits of a generic pointer to __shared__ = wave-relative LDS byte
// offset (flat aperture); exactly what the async
// instruction's VDST operand wants.
__device__ __forceinline__ uint32_t lds_off(const void* p) {
    return (uint32_t)(uintptr_t)p;
}

__device__ __forceinline__ void async_b128(uint32_t lds_dst, const float* gsrc) {
    asm volatile("global_load_async_to_lds_b128 %0, %1, off"
                 :: "v"(lds_dst), "v"(gsrc) : "memory");
}

#define WAIT_ASYNC(n) asm volatile("s_wait_asynccnt %0" :: "n"(n) : "memory")

template<int KSTRIDE, int KLEN, int NCOL, bool SUMA>
__global__ __launch_bounds__(256, 2)
void kc_expert_gemm(const float* __restrict__ A,    // [4096][KSTRIDE]
                    const float* __restrict__ A2,   // second partial (SUMA only)
                    const float* __restrict__ W,    // [4096][NE]
                    const float* __restrict__ Bm,   // [NE][KSTRIDE][NCOL]
                    float*       __restrict__ OutB) // [gridDim.z][4096][NCOL]
{
    __shared__ float sA[MT * ASTR];      //  9216 B
    __shared__ float sW[MT * WSTR];      //  8448 B
    __shared__ float sB[2 * BBUF];       // 34816 B (double buffer)

    const int tid  = threadIdx.x;
    const int lane = tid & 31;
    const int wave = tid >> 5;
    const int wm   = wave >> 2;          // 0..1 : 32-row half (2 x 16-row subtiles)
    const int wn   = wave & 3;           // 0..3 : 32-col quarter (2 x 16-col blocks)
    const int l16  = lane & 15;
    const int lhi  = lane >> 4;          // lane half selects the K pair

    const int row0  = blockIdx.x * MT;
    const int col0  = blockIdx.y * CT;
    const int kbase = blockIdx.z * KLEN; // split-K offset (phase 1)
    float* Out = OutB + (size_t)blockIdx.z * 4096 * NCOL;

    v8f acc00 = {0.f,0.f,0.f,0.f,0.f,0.f,0.f,0.f};
    v8f acc01 = acc00, acc10 = acc00, acc11 = acc00;

    // Stage expert weights once (scalar stores; WSTR=33 keeps reads conflict-free).
    #pragma unroll
    for (int j = 0; j < (MT * NE) / 256; ++j) {
        const int idx = tid + j * 256;
        const int r = idx >> 5, cc = idx & 31;
        sW[r * WSTR + cc] = W[(size_t)(row0 + r) * NE + cc];
    }

    // --- A chunk staging: async b128 (phase 1) or load+sum of split-K partials (phase 2).
    auto stageA = [&](int kc) {
        if (SUMA) {
            #pragma unroll
            for (int j = 0; j < 2; ++j) {
                const int idx = tid + j * 256;            // MT*KC/4 = 512 vec4
                const int r = idx >> 3, c4 = idx & 7;
                const size_t go = (size_t)(row0 + r) * KSTRIDE + kbase + kc * KC + c4 * 4;
                float4 va = *(const float4*)(A + go);
                const float4 vb = *(const float4*)(A2 + go);
                va.x += vb.x; va.y += vb.y; va.z += vb.z; va.w += vb.w;
                *(float4*)&sA[r * ASTR + c4 * 4] = va;
            }
        } else {
            // 64 rows x 128 B; one async instr moves 4 rows (8 lanes/row x 16B).
            #pragma unroll
            for (int j = 0; j < 2; ++j) {
                const int r  = wave * 8 + j * 4 + (lane >> 3);
                const int cf = (lane & 7) * 4;
                async_b128(lds_off(&sA[r * ASTR + cf]),
                           A + (size_t)(row0 + r) * KSTRIDE + kbase + kc * KC + cf);
            }
        }
    };

    // --- B chunk staging: 32 rows x 512 B, one async instr per row, 4 rows/wave.
    auto stageB = [&](int c) {
        const int n  = c & (NE - 1);
        const int kc = c >> 5;
        float* buf = &sB[(c & 1) * BBUF];
        #pragma unroll
        for (int j = 0; j < 4; ++j) {
            const int r = wave * 4 + j;
            async_b128(lds_off(&buf[r * BSTR + lane * 4]),
                       Bm + ((size_t)n * KSTRIDE + kbase + kc * KC + r) * NCOL
                          + col0 + lane * 4);
        }
    };

    constexpr int NC = (KLEN / KC) * NE;  // chunks = K-chunks x experts
    stageA(0);
    stageB(0);

    for (int c = 0; c < NC; ++c) {
        const int n = c & (NE - 1);
        if (n == 0 && c > 0) stageA(c >> 5);          // safe: barrier at end of c-1
        if (c + 1 < NC) { stageB(c + 1); WAIT_ASYNC(4); }  // in-order => chunk c + A landed
        else            { WAIT_ASYNC(0); }
        __syncthreads();                               // staged data visible to all waves

        const float* buf = &sB[(c & 1) * BBUF];
        const float wgt0 = sW[(wm * 32      + l16) * WSTR + n];
        const float wgt1 = sW[(wm * 32 + 16 + l16) * WSTR + n];

        #pragma unroll
        for (int k = 0; k < KC; k += 4) {
            // A 16x4 fragments for both row subtiles, expert weight folded in.
            v2f a0 = *(const v2f*)&sA[(wm * 32      + l16) * ASTR + k + lhi * 2];
            v2f a1 = *(const v2f*)&sA[(wm * 32 + 16 + l16) * ASTR + k + lhi * 2];
            a0 *= wgt0; a1 *= wgt1;
            // B 4x16 fragments: two K-adjacent rows per lane -> ds_load_2addr_b32.
            const float* bp = &buf[(k + lhi * 2) * BSTR + wn * 32 + l16];
            v2f b0; b0.x = bp[0];  b0.y = bp[BSTR];
            v2f b1; b1.x = bp[16]; b1.y = bp[16 + BSTR];
            acc00 = __builtin_amdgcn_wmma_f32_16x16x4_f32(false,a0,false,b0,(short)0,acc00,false,false);
            acc01 = __builtin_amdgcn_wmma_f32_16x16x4_f32(false,a0,false,b1,(short)0,acc01,false,false);
            acc10 = __builtin_amdgcn_wmma_f32_16x16x4_f32(false,a1,false,b0,(short)0,acc10,false,false);
            acc11 = __builtin_amdgcn_wmma_f32_16x16x4_f32(false,a1,false,b1,(short)0,acc11,false,false);
        }
        __syncthreads();           // all readers done before next overwrite of sA/sB
    }

    // C/D layout: VGPR v, lane l -> row = v + (l>=16)*8, col = l&15.
    const int orow = row0 + wm * 32 + lhi * 8;
    const int ocol = col0 + wn * 32 + l16;
    #pragma unroll
    for (int v = 0; v < 8; ++v) {
        Out[(size_t)(orow + v)      * NCOL + ocol     ] = acc00[v];
        Out[(size_t)(orow + v)      * NCOL + ocol + 16] = acc01[v];
        Out[(size_t)(orow + 16 + v) * NCOL + ocol     ] = acc10[v];
        Out[(size_t)(orow + 16 + v) * NCOL + ocol + 16] = acc11[v];
    }
}

extern "C" void kernel_launch(void* const* d_in, const int* in_sizes, int n_in,
                              void* d_out, int out_size, void* d_ws, size_t ws_size,
                              hipStream_t stream) {
    (void)in_sizes; (void)n_in; (void)out_size; (void)ws_size;
    const float* x  = (const float*)d_in[0];  // (4,1024,1024)
    const float* fw = (const float*)d_in[1];  // (4,1024,32)
    const float* rw = (const float*)d_in[2];  // (4,1024,32)
    const float* FK = (const float*)d_in[3];  // (32,1024,128)
    const float* RK = (const float*)d_in[4];  // (32,128,1024)
    float* out = (float*)d_out;               // (4,1024,1024)
    float* h0  = (float*)d_ws;                // two (4096,128) split-K partials (4MB)
    float* h1  = h0 + (size_t)4096 * 128;

    dim3 blk(256);
    // Phase 1: split-K over z -> h0 (z=0) and h1 (z=1); 128 blocks.
    kc_expert_gemm<1024, 512, 128, false><<<dim3(64, 1, 2), blk, 0, stream>>>(x, x, fw, FK, h0);
    // Phase 2: A = h0 + h1 (summed in A-staging); 512 blocks.
    kc_expert_gemm<128, 128, 1024, true><<<dim3(64, 8, 1), blk, 0, stream>>>(h0, h1, rw, RK, out);
}